// KernelActivation_46763603918994
// MI455X (gfx1250) — compile-verified
//
#include <hip/hip_runtime.h>

// Pure permutation: out[((b*Ci+i)*Hj+j)*W*4 + w*4 + p*2 + q] = x[b, 2i+p, 2j+q, w]
// x: (16, 64, 224, 224) f32.  Bandwidth-bound: 411 MB total, ~17.6 us floor @ 23.3 TB/s.
// Single-use streaming data (411 MB > 192 MB L2) -> non-temporal TH hints both ways.
//
// Grid mapping carries all tile indices in blockIdx (zero per-thread divisions):
//   grid = (Hj, Ci, Bn), block = (224) = 7 full wave32s = exactly one tile row set.
// Every load/store base is block-uniform (SGPR + saddr addressing); per-lane
// offset is just threadIdx.x scaled.

namespace {
constexpr unsigned Bn = 16, C = 64, H = 224, W = 224;
constexpr unsigned Kc = 2;
constexpr unsigned Ci = C / Kc;   // 32
constexpr unsigned Hj = H / Kc;   // 112
}

typedef float float4v __attribute__((ext_vector_type(4)));

__global__ __launch_bounds__(224) void repatch_permute_kernel(
    const float* __restrict__ x, float* __restrict__ out) {
  const unsigned w = threadIdx.x;   // 0..223  (wave-contiguous)
  const unsigned j = blockIdx.x;    // 0..111
  const unsigned i = blockIdx.y;    // 0..31
  const unsigned b = blockIdx.z;    // 0..15

  const size_t c0 = (size_t)b * C + 2u * i;   // global channel of p=0
  const size_t h0 = 2u * j;                   // global row of q=0

  const float* p00 = x + ((c0 + 0u) * H + (h0 + 0u)) * W + w;
  const float* p01 = x + ((c0 + 0u) * H + (h0 + 1u)) * W + w;
  const float* p10 = x + ((c0 + 1u) * H + (h0 + 0u)) * W + w;
  const float* p11 = x + ((c0 + 1u) * H + (h0 + 1u)) * W + w;

  // 4 streaming loads: each is 32 lanes x 4B = 128B contiguous, th:NT
  const float f00 = __builtin_nontemporal_load(p00);
  const float f01 = __builtin_nontemporal_load(p01);
  const float f10 = __builtin_nontemporal_load(p10);
  const float f11 = __builtin_nontemporal_load(p11);

  // One streaming store: 32 lanes x 16B = 512B contiguous, th:NT
  float4v v = {f00, f01, f10, f11};   // order (p,q) = (0,0),(0,1),(1,0),(1,1)
  const size_t tile = ((size_t)b * Ci + i) * Hj + j;
  float4v* dst = (float4v*)(out + tile * (W * 4u) + (size_t)w * 4u);
  __builtin_nontemporal_store(v, dst);
}

extern "C" void kernel_launch(void* const* d_in, const int* in_sizes, int n_in,
                              void* d_out, int out_size, void* d_ws, size_t ws_size,
                              hipStream_t stream) {
  const float* x = (const float*)d_in[0];
  float* out = (float*)d_out;
  (void)in_sizes; (void)n_in; (void)out_size; (void)d_ws; (void)ws_size;

  dim3 block(224);          // 7 wave32s, exactly one tile width
  dim3 grid(Hj, Ci, Bn);    // (112, 32, 16) = 57344 tiles
  repatch_permute_kernel<<<grid, block, 0, stream>>>(x, out);
}